// SinkhornRouter_23596550324422
// MI455X (gfx1250) — compile-verified
//
#include <hip/hip_runtime.h>

#define B_ 16
#define N_ 8192
#define K_ 256
#define CSTRIDE 260          // per-batch stride for C/S (1040B, 16B aligned)
#define LP 272               // LDS row stride (floats)
#define EPSV 1e-8f
#define ROW_T (1.0f/8192.0f)
#define COL_MAIN (0.9f/256.0f)
#define COL_LAST 0.1f
#define SCALE_L2E 28.853900817779268f   // (1/0.05) * log2(e)

typedef __attribute__((ext_vector_type(2))) float v2f;
typedef __attribute__((ext_vector_type(8))) float v8f;

__global__ void sk_init(float* __restrict__ R, float* __restrict__ C, float* __restrict__ S) {
    int i = blockIdx.x * 256 + threadIdx.x;
    if (i < B_ * N_) R[i] = 1.0f;
    if (i < B_ * CSTRIDE) {
        int k = i % CSTRIDE;
        C[i] = (k <= K_) ? 1.0f : 0.0f;
        S[i] = 0.0f;
    }
}

// One fused Sinkhorn iteration: row-normalize (update R) and accumulate
// column partial sums S[k] = sum_n E[n][k] * R_new[n] (without C factor).
__global__ __launch_bounds__(256) void sk_iter(const float* __restrict__ in,
                                               float* __restrict__ R,
                                               const float* __restrict__ C,
                                               float* __restrict__ S) {
    __shared__ float P[8 * LP];
    const int lane = threadIdx.x & 31;
    const int wave = threadIdx.x >> 5;
    const int b = blockIdx.y;

    for (int i = threadIdx.x; i < 8 * LP; i += 256) P[i] = 0.0f;

    // Per-lane C registers: lane l owns cols 4l..4l+3 and 128+4l..128+4l+3.
    const float4* Cb = (const float4*)(C + b * CSTRIDE);
    const float4 cA = Cb[lane];
    const float4 cB = Cb[32 + lane];
    const float clast = C[b * CSTRIDE + 256];

    float a0=0,a1=0,a2=0,a3=0,a4=0,a5=0,a6=0,a7=0, wsum=0;
    const size_t rowbase = (size_t)b * N_ + (size_t)blockIdx.x * 256;

    for (int i = 0; i < 32; ++i) {
        const size_t g = rowbase + i * 8 + wave;
        const float4* rp = (const float4*)(in + g * (size_t)K_);
        if (i + 2 < 32)   // prefetch 2 row-steps ahead (first HBM pass)
            __builtin_prefetch(in + (g + 16) * (size_t)K_ + 4 * lane, 0, 3);
        const float4 sA = rp[lane];
        const float4 sB = rp[32 + lane];
        const float e0 = __builtin_amdgcn_exp2f(sA.x * SCALE_L2E);
        const float e1 = __builtin_amdgcn_exp2f(sA.y * SCALE_L2E);
        const float e2 = __builtin_amdgcn_exp2f(sA.z * SCALE_L2E);
        const float e3 = __builtin_amdgcn_exp2f(sA.w * SCALE_L2E);
        const float e4 = __builtin_amdgcn_exp2f(sB.x * SCALE_L2E);
        const float e5 = __builtin_amdgcn_exp2f(sB.y * SCALE_L2E);
        const float e6 = __builtin_amdgcn_exp2f(sB.z * SCALE_L2E);
        const float e7 = __builtin_amdgcn_exp2f(sB.w * SCALE_L2E);
        float u = e0*cA.x + e1*cA.y + e2*cA.z + e3*cA.w
                + e4*cB.x + e5*cB.y + e6*cB.z + e7*cB.w;
        for (int m = 16; m >= 1; m >>= 1) u += __shfl_xor(u, m, 32);
        u += clast;                              // last column: E = 1
        const float Rold = R[g];
        const float Rnew = Rold * ROW_T / (Rold * u + EPSV);
        if (lane == 0) R[g] = Rnew;
        a0 += e0*Rnew; a1 += e1*Rnew; a2 += e2*Rnew; a3 += e3*Rnew;
        a4 += e4*Rnew; a5 += e5*Rnew; a6 += e6*Rnew; a7 += e7*Rnew;
        wsum += Rnew;
    }

    float* Pw = P + wave * LP;
    Pw[4*lane + 0] = a0; Pw[4*lane + 1] = a1; Pw[4*lane + 2] = a2; Pw[4*lane + 3] = a3;
    Pw[128 + 4*lane + 0] = a4; Pw[128 + 4*lane + 1] = a5;
    Pw[128 + 4*lane + 2] = a6; Pw[128 + 4*lane + 3] = a7;
    if (lane == 0) Pw[256] = wsum;
    __syncthreads();

    // Wave 0 reduces the 8x272 partial tile with V_WMMA_F32_16X16X4_F32:
    // A = ones(16x4)  =>  D[m][n] = sum_k B[k][n]  (column sums).
    if (wave == 0) {
        const int nn = lane & 15;
        const int kb = (lane >= 16) ? 2 : 0;   // B layout: VGPR0 holds K=0 (lanes 0-15) / K=2 (16-31)
        v2f A; A.x = 1.0f; A.y = 1.0f;
        for (int t = 0; t < 17; ++t) {
            const int base = t * 16;
            v2f Blo, Bhi;
            Blo.x = P[(kb + 0) * LP + base + nn];  // waves 0..3
            Blo.y = P[(kb + 1) * LP + base + nn];
            Bhi.x = P[(kb + 4) * LP + base + nn];  // waves 4..7
            Bhi.y = P[(kb + 5) * LP + base + nn];
            v8f d = {};
            d = __builtin_amdgcn_wmma_f32_16x16x4_f32(false, A, false, Blo, (short)0, d, false, false);
            d = __builtin_amdgcn_wmma_f32_16x16x4_f32(false, A, false, Bhi, (short)0, d, false, false);
            const int col = base + nn;
            if (lane < 16 && col <= K_)
                atomicAdd(&S[b * CSTRIDE + col], d[0]);   // D[0][nn] in VGPR0, lanes 0-15
        }
    }
}

// Finish column normalization: C_k <- C_k * tgt_k / (C_k * S_k + eps); reset S.
__global__ void sk_colfix(float* __restrict__ C, float* __restrict__ S) {
    const int b = blockIdx.x;
    const int t = threadIdx.x;
    if (t <= K_) {
        const float c = C[b * CSTRIDE + t];
        const float s = S[b * CSTRIDE + t];
        const float tgt = (t < K_) ? COL_MAIN : COL_LAST;
        C[b * CSTRIDE + t] = c * tgt / (c * s + EPSV);
        S[b * CSTRIDE + t] = 0.0f;
    }
}

// Final pass: q <- q / (rowsum + eps); write assign and confidence.
__global__ __launch_bounds__(256) void sk_final(const float* __restrict__ in,
                                                const float* __restrict__ R,
                                                const float* __restrict__ C,
                                                float* __restrict__ assign,
                                                float* __restrict__ conf) {
    const int lane = threadIdx.x & 31;
    const int wave = threadIdx.x >> 5;
    const int b = blockIdx.y;
    const float4* Cb = (const float4*)(C + b * CSTRIDE);
    const float4 cA = Cb[lane];
    const float4 cB = Cb[32 + lane];
    const float clast = C[b * CSTRIDE + 256];
    const size_t rowbase = (size_t)b * N_ + (size_t)blockIdx.x * 256;

    for (int i = 0; i < 32; ++i) {
        const size_t g = rowbase + i * 8 + wave;
        const float4* rp = (const float4*)(in + g * (size_t)K_);
        if (i + 2 < 32)
            __builtin_prefetch(in + (g + 16) * (size_t)K_ + 4 * lane, 0, 3);
        const float4 sA = rp[lane];
        const float4 sB = rp[32 + lane];
        const float e0 = __builtin_amdgcn_exp2f(sA.x * SCALE_L2E);
        const float e1 = __builtin_amdgcn_exp2f(sA.y * SCALE_L2E);
        const float e2 = __builtin_amdgcn_exp2f(sA.z * SCALE_L2E);
        const float e3 = __builtin_amdgcn_exp2f(sA.w * SCALE_L2E);
        const float e4 = __builtin_amdgcn_exp2f(sB.x * SCALE_L2E);
        const float e5 = __builtin_amdgcn_exp2f(sB.y * SCALE_L2E);
        const float e6 = __builtin_amdgcn_exp2f(sB.z * SCALE_L2E);
        const float e7 = __builtin_amdgcn_exp2f(sB.w * SCALE_L2E);
        float u = e0*cA.x + e1*cA.y + e2*cA.z + e3*cA.w
                + e4*cB.x + e5*cB.y + e6*cB.z + e7*cB.w;
        for (int m = 16; m >= 1; m >>= 1) u += __shfl_xor(u, m, 32);
        u += clast;
        const float Rold = R[g];
        const float tt = Rold / (Rold * u + EPSV);
        float4 oA, oB;
        oA.x = e0*cA.x*tt; oA.y = e1*cA.y*tt; oA.z = e2*cA.z*tt; oA.w = e3*cA.w*tt;
        oB.x = e4*cB.x*tt; oB.y = e5*cB.y*tt; oB.z = e6*cB.z*tt; oB.w = e7*cB.w*tt;
        float4* op = (float4*)(assign + g * (size_t)K_);
        op[lane] = oA;
        op[32 + lane] = oB;
        if (lane == 0) conf[g] = 1.0f - clast * tt;
    }
}

extern "C" void kernel_launch(void* const* d_in, const int* in_sizes, int n_in,
                              void* d_out, int out_size, void* d_ws, size_t ws_size,
                              hipStream_t stream) {
    const float* in = (const float*)d_in[0];
    float* ws = (float*)d_ws;
    float* R = ws;                       // 16*8192 floats
    float* C = R + B_ * N_;              // 16*260 floats
    float* S = C + B_ * CSTRIDE;         // 16*260 floats
    float* assign = (float*)d_out;
    float* conf = assign + (size_t)B_ * N_ * K_;

    sk_init<<<512, 256, 0, stream>>>(R, C, S);
    for (int it = 0; it < 5; ++it) {
        sk_iter<<<dim3(32, 16), 256, 0, stream>>>(in, R, C, S);
        sk_colfix<<<16, 288, 0, stream>>>(C, S);
    }
    sk_final<<<dim3(32, 16), 256, 0, stream>>>(in, R, C, assign, conf);
}